// GlobalGATLayer_88441966559845
// MI455X (gfx1250) — compile-verified
//
#include <hip/hip_runtime.h>

// ---------------------------------------------------------------------------
// GAT layer for MI455X (gfx1250, wave32).
// Constants from reference: N=50000, IN_DIM=128, HEADS=4, OUT_DIM=32, H*D=128.
// ---------------------------------------------------------------------------

typedef __attribute__((ext_vector_type(2))) float v2f;
typedef __attribute__((ext_vector_type(8))) float v8f;

#define IN_DIM   128
#define HD       128   // HEADS * OUT_DIM
#define HEADS    4
#define ODIM     32
#define NEG_SLOPE 0.2f

// ---------------------------------------------------------------------------
// Zero a region of workspace (acc + s + m_u).
// ---------------------------------------------------------------------------
__global__ void gat_zero_kernel(float* __restrict__ p, unsigned long n) {
    unsigned long i = (unsigned long)blockIdx.x * blockDim.x + threadIdx.x;
    unsigned long stride = (unsigned long)gridDim.x * blockDim.x;
    for (; i < n; i += stride) p[i] = 0.0f;
}

// ---------------------------------------------------------------------------
// h = x @ W  via V_WMMA_F32_16X16X4_F32.
// Block = 256 threads = 8 waves. Each block computes a 16x128 tile of h:
// wave w computes output columns [16w, 16w+16). K loop: 128/4 = 32 WMMAs.
//
// fp32 WMMA fragment layout (ISA 7.12.2):
//   A 16x4 : lane%16 = M row, lane/16 selects K half; VGPR v holds K = 2*(lane/16)+v
//   B 4x16 : lane%16 = N col, VGPR v holds K = 2*(lane/16)+v
//   C/D    : lane%16 = N col, VGPR v holds M = v + 8*(lane/16)
// ---------------------------------------------------------------------------
__global__ __launch_bounds__(256) void gat_gemm_wmma(
    const float* __restrict__ x, const float* __restrict__ W,
    float* __restrict__ h, int N)
{
    const int wave = threadIdx.x >> 5;   // 0..7 -> column tile
    const int lane = threadIdx.x & 31;
    const int idx  = lane & 15;          // M row (for A) / N col (for B,C,D)
    const int grp  = lane >> 4;          // K half selector
    const int rowBase = blockIdx.x * 16;
    if (rowBase >= N) return;

    const float* __restrict__ xrow = x + (size_t)(rowBase + idx) * IN_DIM;
    const int n = wave * 16 + idx;       // output column

    v8f c = {};
#pragma unroll
    for (int k = 0; k < IN_DIM; k += 4) {
        const int kk = k + 2 * grp;
        // A fragment: 2 contiguous K values of this M row -> single b64 load
        v2f a = *(const v2f*)(xrow + kk);
        // B fragment: W[kk][n], W[kk+1][n] (row-major W[IN_DIM][HD])
        v2f b;
        b.x = W[(size_t)kk * HD + n];
        b.y = W[(size_t)(kk + 1) * HD + n];
        c = __builtin_amdgcn_wmma_f32_16x16x4_f32(
                /*neg_a=*/false, a, /*neg_b=*/false, b,
                /*c_mod=*/(short)0, c, /*reuse_a=*/false, /*reuse_b=*/false);
    }

#pragma unroll
    for (int v = 0; v < 8; ++v) {
        h[(size_t)(rowBase + v + 8 * grp) * HD + n] = c[v];
    }
}

// Monotonic float <-> uint mapping so GLOBAL_ATOMIC_MAX_U32 implements fmax.
__device__ __forceinline__ unsigned fmap(float x) {
    unsigned u = __float_as_uint(x);
    return (u & 0x80000000u) ? ~u : (u | 0x80000000u);
}
__device__ __forceinline__ float funmap(unsigned u) {
    return __uint_as_float((u & 0x80000000u) ? (u & 0x7FFFFFFFu) : ~u);
}

// ---------------------------------------------------------------------------
// Pass A: one wave32 per (edge, head); lane = feature dim d in [0,32).
//   score = <h[src], a_src> + <h[dst], a_dst>  (over D=32, wave reduction)
//   e[edge,head] = leaky_relu(score); atomic-umax into m_u[dst,head].
// All gathers are coalesced 128B transactions that hit L2 (h = 25.6MB << 192MB).
// ---------------------------------------------------------------------------
__global__ __launch_bounds__(256) void gat_edge_scores(
    const float* __restrict__ h, const int* __restrict__ ei,
    const float* __restrict__ a_src, const float* __restrict__ a_dst,
    float* __restrict__ e, unsigned* __restrict__ m_u, int E)
{
    const int waveId = blockIdx.x * 8 + (threadIdx.x >> 5);
    const int lane   = threadIdx.x & 31;
    const int edge   = waveId >> 2;
    const int head   = waveId & 3;
    if (edge >= E) return;

    const int sn = ei[edge];
    const int dn = ei[E + edge];
    const int off = head * ODIM + lane;

    float v = h[(size_t)sn * HD + off] * a_src[off]
            + h[(size_t)dn * HD + off] * a_dst[off];
#pragma unroll
    for (int m = 16; m >= 1; m >>= 1) v += __shfl_xor(v, m, 32);

    if (lane == 0) {
        float lv = v > 0.0f ? v : NEG_SLOPE * v;
        e[(size_t)edge * HEADS + head] = lv;
        atomicMax(&m_u[(size_t)dn * HEADS + head], fmap(lv));
    }
}

// ---------------------------------------------------------------------------
// Pass B: one wave32 per (edge, head); lane = feature dim.
//   ex = exp(e - m[dst]); acc[dst,head,:] += ex * h[src,head,:]; s[dst,head] += ex
// Normalization by s is deferred to the finalize kernel (alpha = ex/s factors out
// of the segment sum), saving a third pass over the edges.
// ---------------------------------------------------------------------------
__global__ __launch_bounds__(256) void gat_edge_accum(
    const float* __restrict__ h, const int* __restrict__ ei,
    const float* __restrict__ e, const unsigned* __restrict__ m_u,
    float* __restrict__ acc, float* __restrict__ s, int E)
{
    const int waveId = blockIdx.x * 8 + (threadIdx.x >> 5);
    const int lane   = threadIdx.x & 31;
    const int edge   = waveId >> 2;
    const int head   = waveId & 3;
    if (edge >= E) return;

    const int sn = ei[edge];
    const int dn = ei[E + edge];

    const float mv = funmap(m_u[(size_t)dn * HEADS + head]);
    const float ex = __expf(e[(size_t)edge * HEADS + head] - mv);

    const int off = head * ODIM + lane;
    const float hv = h[(size_t)sn * HD + off];
    atomicAdd(&acc[(size_t)dn * HD + off], ex * hv);
    if (lane == 0) atomicAdd(&s[(size_t)dn * HEADS + head], ex);
}

// ---------------------------------------------------------------------------
// Finalize: out[n,d] = mean_h acc[n,h,d] / (s[n,h] + 1e-12)
// ---------------------------------------------------------------------------
__global__ __launch_bounds__(256) void gat_finalize(
    const float* __restrict__ acc, const float* __restrict__ s,
    float* __restrict__ out, int N)
{
    const int t = blockIdx.x * blockDim.x + threadIdx.x;
    if (t >= N * ODIM) return;
    const int n = t >> 5;      // node
    const int d = t & 31;      // out dim
    float r = 0.0f;
#pragma unroll
    for (int hh = 0; hh < HEADS; ++hh) {
        r += acc[(size_t)n * HD + hh * ODIM + d] /
             (s[(size_t)n * HEADS + hh] + 1e-12f);
    }
    out[t] = r * (1.0f / HEADS);
}

// ---------------------------------------------------------------------------
// Launch. Workspace layout (floats):
//   [ h : N*128 ][ acc : N*128 ][ s : N*4 ][ m_u : N*4 (u32) ][ e : E*4 ]
// acc/s/m_u are contiguous -> single zero-fill of N*136 words.
// ---------------------------------------------------------------------------
extern "C" void kernel_launch(void* const* d_in, const int* in_sizes, int n_in,
                              void* d_out, int out_size, void* d_ws, size_t ws_size,
                              hipStream_t stream) {
    const float* x     = (const float*)d_in[0];
    const int*   ei    = (const int*)d_in[1];
    const float* W     = (const float*)d_in[2];
    const float* a_src = (const float*)d_in[3];
    const float* a_dst = (const float*)d_in[4];

    const int N = in_sizes[0] / IN_DIM;
    const int E = in_sizes[1] / 2;

    float* ws  = (float*)d_ws;
    float* h   = ws;
    float* acc = h   + (size_t)N * HD;
    float* s   = acc + (size_t)N * HD;
    unsigned* m_u = (unsigned*)(s + (size_t)N * HEADS);
    float* e   = (float*)(m_u + (size_t)N * HEADS);

    // 1) zero acc + s + m_u (contiguous region)
    const unsigned long zwords = (unsigned long)N * (HD + 2 * HEADS);
    gat_zero_kernel<<<4096, 256, 0, stream>>>(acc, zwords);

    // 2) h = x @ W  (fp32 WMMA)
    gat_gemm_wmma<<<(N + 15) / 16, 256, 0, stream>>>(x, W, h, N);

    // 3) edge scores + segment max
    const long pairs = (long)E * HEADS;           // wave32 per (edge, head)
    const int blocksE = (int)((pairs + 7) / 8);   // 8 waves per block
    gat_edge_scores<<<blocksE, 256, 0, stream>>>(h, ei, a_src, a_dst, e, m_u, E);

    // 4) exp + scatter-accumulate
    gat_edge_accum<<<blocksE, 256, 0, stream>>>(h, ei, e, m_u, acc, s, E);

    // 5) normalize + head mean
    gat_finalize<<<(N * ODIM + 255) / 256, 256, 0, stream>>>(acc, s, (float*)d_out, N);
}